// ConvFace_11441792876787
// MI455X (gfx1250) — compile-verified
//
#include <hip/hip_runtime.h>
#include <hip/hip_bf16.h>

#define M_  8
#define C_  128
#define O_  256
#define F_  16384
#define FP_ 8192
#define K_  16
#define BN_EPS 1e-5f
#define LDS_STRIDE 72   // 64 data dwords + 8 pad dwords per row (set by TDM pad fields)

typedef __attribute__((ext_vector_type(2))) float v2f;
typedef __attribute__((ext_vector_type(8))) float v8f;
typedef __attribute__((ext_vector_type(4))) unsigned int v4u;
typedef __attribute__((ext_vector_type(4))) int v4i;
typedef __attribute__((ext_vector_type(8))) int v8i;

// ---------------------------------------------------------------------------
// Kernel 1: gather + neighbor sum.  x[m][c][fp] = fea[m][c][pool] + sum_k fea[m][c][ring]
// grid: (FP_/256, M_), block: 256
// ---------------------------------------------------------------------------
__global__ __launch_bounds__(256) void gather_sum_kernel(
    const float* __restrict__ fea,
    const long long* __restrict__ ring_n,
    const long long* __restrict__ pool_idx,
    float* __restrict__ x)
{
    const int m  = blockIdx.y;
    const int fp = blockIdx.x * 256 + threadIdx.x;

    int idx[K_ + 1];
    const long long* rn = ring_n + ((long long)m * FP_ + fp) * K_;
#pragma unroll
    for (int k = 0; k < K_; ++k) idx[k] = (int)rn[k];
    idx[K_] = (int)pool_idx[fp];

    const float* __restrict__ fm = fea + (long long)m * C_ * F_;
    float* __restrict__ xm = x + (long long)m * C_ * FP_;

    for (int c = 0; c < C_; ++c) {
        const float* __restrict__ row = fm + (long long)c * F_;
        float acc = row[idx[K_]];
#pragma unroll
        for (int k = 0; k < K_; ++k) acc += row[idx[k]];
        xm[(long long)c * FP_ + fp] = acc;
    }
}

// ---------------------------------------------------------------------------
// Kernel 2: y[m][o][fp] = sum_c W[o][c]*x[m][c][fp] + b[o]   (WMMA f32 16x16x4)
// grid: (FP_/64, M_), block: 256 (8 waves).
// Wave 0 TDM-loads x[m][0:128][fp0:fp0+64] into LDS (row stride 72 dwords via
// hardware padding); each wave then computes 2 o-tiles x 4 fp-tiles.
// ---------------------------------------------------------------------------
__global__ __launch_bounds__(256) void gemm_wmma_kernel(
    const float* __restrict__ W,
    const float* __restrict__ bias,
    const float* __restrict__ x,
    float* __restrict__ y)
{
    __shared__ float tile[C_ * LDS_STRIDE];          // 36864 bytes

    const int m    = blockIdx.y;
    const int fp0  = blockIdx.x * 64;
    const int wave = threadIdx.x >> 5;               // 0..7
    const int lane = threadIdx.x & 31;               // wave32
    const int col  = lane & 15;
    const int half = lane >> 4;

    // AA defeat #1: an opaque never-taken store so LLVM cannot prove tile[] is
    // write-free and fold its loads to undef (round-2 failure mode).
    unsigned guard;
    asm volatile("s_mov_b32 %0, 0" : "=s"(guard));
    if (guard) tile[threadIdx.x] = 0.0f;

    // LDS byte offset of tile[]: flat-aperture LDS addresses carry the LDS
    // offset in the low 32 bits (ISA 10.2). This also escapes tile[] (AA #2).
    const unsigned lds_off = (unsigned)(unsigned long long)(void*)tile;

    // ---- Tensor Data Mover: one DMA per block, issued by wave 0 only --------
    if (wave == 0) {
        const unsigned long long ga =
            (unsigned long long)(x + ((long long)m * C_ * FP_ + fp0));
        v4u g0;
        g0[0] = 1u;                                            // count=1, user mode
        g0[1] = lds_off;                                       // lds_addr (bytes)
        g0[2] = (unsigned)(ga & 0xffffffffull);                // global_addr[31:0]
        g0[3] = (unsigned)((ga >> 32) & 0x01ffffffull)         // global_addr[56:32]
              | (2u << 30);                                    // type = 2 (image)
        v8i g1;
        g1[0] = (int)((2u << 16)      // data_size = 4 bytes
                    | (1u << 20)      // pad_enable
                    | (5u << 22)      // pad_interval: 64 dwords
                    | (7u << 25));    // pad_amount:   8 dwords -> LDS row stride 72
        g1[1] = (int)(((unsigned)FP_ & 0xffffu) << 16);        // tensor_dim0[15:0]
        g1[2] = (int)(((unsigned)C_  & 0xffffu) << 16);        // tensor_dim1[15:0]
        g1[3] = (int)(64u << 16);                              // tile_dim0 = 64
        g1[4] = (int)C_;                                       // tile_dim1 = 128, tile_dim2 = 0
        g1[5] = (int)FP_;                                      // tensor_dim0_stride[31:0]
        g1[6] = 0;                                             // stride hi / dim1_stride lo
        g1[7] = 0;                                             // dim1_stride hi
        v4i gz = {0, 0, 0, 0};
#if __has_include(<hip/amd_detail/amd_gfx1250_TDM.h>)
        v8i gz8 = {0, 0, 0, 0, 0, 0, 0, 0};
        __builtin_amdgcn_tensor_load_to_lds(g0, g1, gz, gz, gz8, 0);
#else
        __builtin_amdgcn_tensor_load_to_lds(g0, g1, gz, gz, 0);
#endif
        __builtin_amdgcn_s_wait_tensorcnt(0);
    }
    // AA defeat #3: tile[] has escaped above, so a generic memory clobber must
    // be assumed to (re)write it -> loads below must really read LDS.
    asm volatile("" ::: "memory");
    __syncthreads();

    const int o0 = wave * 32;
    const int o1 = o0 + 16;

    v8f acc0[4] = {};
    v8f acc1[4] = {};

#pragma unroll 2
    for (int kk = 0; kk < C_ / 4; ++kk) {
        const int kh = kk * 4 + 2 * half;            // K pair for this half-wave

        // A tiles: a[j] = W[o + col][kh + j]  (contiguous -> one v2f load each)
        const v2f a0 = *(const v2f*)(W + (long long)(o0 + col) * C_ + kh);
        const v2f a1 = *(const v2f*)(W + (long long)(o1 + col) * C_ + kh);

#pragma unroll
        for (int t = 0; t < 4; ++t) {
            v2f b;
            b.x = tile[(kh + 0) * LDS_STRIDE + t * 16 + col];
            b.y = tile[(kh + 1) * LDS_STRIDE + t * 16 + col];
            acc0[t] = __builtin_amdgcn_wmma_f32_16x16x4_f32(
                false, a0, false, b, (short)0, acc0[t], false, false);
            acc1[t] = __builtin_amdgcn_wmma_f32_16x16x4_f32(
                false, a1, false, b, (short)0, acc1[t], false, false);
        }
    }

    // D layout: VGPR r holds row (r + 8*half), col = lane%16
    float* __restrict__ ym = y + (long long)m * O_ * FP_;
#pragma unroll
    for (int t = 0; t < 4; ++t) {
        const int fpc = fp0 + t * 16 + col;
#pragma unroll
        for (int r = 0; r < 8; ++r) {
            const int oa = o0 + r + 8 * half;
            const int ob = o1 + r + 8 * half;
            ym[(long long)oa * FP_ + fpc] = acc0[t][r] + bias[oa];
            ym[(long long)ob * FP_ + fpc] = acc1[t][r] + bias[ob];
        }
    }
}

// ---------------------------------------------------------------------------
// Kernel 3: per-channel batch stats -> fused scale/shift.
// ---------------------------------------------------------------------------
__global__ __launch_bounds__(256) void bn_stats_kernel(
    const float* __restrict__ y,
    const float* __restrict__ gamma,
    const float* __restrict__ beta,
    float* __restrict__ scale,
    float* __restrict__ shift)
{
    const int o   = blockIdx.x;
    const int tid = threadIdx.x;
    const int N   = M_ * FP_;

    float s = 0.f, ss = 0.f;
    for (int i = tid; i < N; i += 256) {
        const int m  = i / FP_;
        const int fp = i - m * FP_;
        const float v = y[((long long)m * O_ + o) * FP_ + fp];
        s  += v;
        ss += v * v;
    }

    __shared__ float rs[256];
    __shared__ float rq[256];
    rs[tid] = s;
    rq[tid] = ss;
    __syncthreads();
    for (int step = 128; step > 0; step >>= 1) {
        if (tid < step) {
            rs[tid] += rs[tid + step];
            rq[tid] += rq[tid + step];
        }
        __syncthreads();
    }

    if (tid == 0) {
        const float inv_n = 1.0f / (float)N;
        const float mean  = rs[0] * inv_n;
        const float var   = rq[0] * inv_n - mean * mean;   // biased variance
        const float rstd  = rsqrtf(var + BN_EPS);
        const float sc    = gamma[o] * rstd;
        scale[o] = sc;
        shift[o] = beta[o] - mean * sc;
    }
}

// ---------------------------------------------------------------------------
// Kernel 4: out = relu(y * scale[o] + shift[o])
// ---------------------------------------------------------------------------
__global__ __launch_bounds__(256) void bn_apply_kernel(
    const float* __restrict__ y,
    const float* __restrict__ scale,
    const float* __restrict__ shift,
    float* __restrict__ out)
{
    const long long i = (long long)blockIdx.x * 256 + threadIdx.x;
    const int o = (int)((i / FP_) % O_);
    const float v = y[i] * scale[o] + shift[o];
    out[i] = v > 0.f ? v : 0.f;
}

// ---------------------------------------------------------------------------
extern "C" void kernel_launch(void* const* d_in, const int* in_sizes, int n_in,
                              void* d_out, int out_size, void* d_ws, size_t ws_size,
                              hipStream_t stream)
{
    const float*     fea      = (const float*)d_in[0];
    const float*     W        = (const float*)d_in[1];
    const float*     bias     = (const float*)d_in[2];
    const float*     gamma    = (const float*)d_in[3];
    const float*     beta     = (const float*)d_in[4];
    const long long* ring_n   = (const long long*)d_in[5];
    const long long* pool_idx = (const long long*)d_in[6];

    float* x     = (float*)d_ws;                         // M*C*FP  (33.5 MB)
    float* y     = x + (size_t)M_ * C_ * FP_;            // M*O*FP  (67 MB)
    float* scale = y + (size_t)M_ * O_ * FP_;            // O
    float* shift = scale + O_;                           // O
    float* out   = (float*)d_out;

    {   // 1. gather + neighbor sum
        dim3 grid(FP_ / 256, M_);
        gather_sum_kernel<<<grid, 256, 0, stream>>>(fea, ring_n, pool_idx, x);
    }
    {   // 2. GEMM via WMMA f32 16x16x4 with TDM-staged B tiles
        dim3 grid(FP_ / 64, M_);
        gemm_wmma_kernel<<<grid, 256, 0, stream>>>(W, bias, x, y);
    }
    {   // 3. batch stats -> scale/shift
        bn_stats_kernel<<<O_, 256, 0, stream>>>(y, gamma, beta, scale, shift);
    }
    {   // 4. normalize + relu
        const long long total = (long long)M_ * O_ * FP_;
        bn_apply_kernel<<<(int)(total / 256), 256, 0, stream>>>(y, scale, shift, out);
    }
}